// SEBlock_57604101374733
// MI455X (gfx1250) — compile-verified
//
#include <hip/hip_runtime.h>
#include <hip/hip_bf16.h>

#define B_    32
#define C_    512
#define HW_   3136     // 56*56
#define KTOP_ 409
#define KLOW_ 103
#define MP_   112      // padded M (7 tiles of 16)
#define KP_   104      // padded K (26 steps of 4)
#define NCH_  112      // pixels per block chunk; 3136 = 28 * 112 exactly
#define KSTEPS_ 26

typedef float v2f __attribute__((ext_vector_type(2)));
typedef float v8f __attribute__((ext_vector_type(8)));

// ---------------------------------------------------------------------------
// Kernel A: global average pool  pooled[b*C+c] = mean(x[b,c,:,:])
// ---------------------------------------------------------------------------
__global__ void se_pool_kernel(const float* __restrict__ x, float* __restrict__ pooled) {
    __shared__ float red[256];
    const int bc = blockIdx.x;                       // b*C_ + c
    const float* row = x + (size_t)bc * HW_;
    float s = 0.f;
    for (int i = threadIdx.x; i < HW_; i += 256) s += row[i];
    red[threadIdx.x] = s;
    __syncthreads();
    for (int off = 128; off > 0; off >>= 1) {
        if ((int)threadIdx.x < off) red[threadIdx.x] += red[threadIdx.x + off];
        __syncthreads();
    }
    if (threadIdx.x == 0) pooled[bc] = red[0] * (1.0f / (float)HW_);
}

// ---------------------------------------------------------------------------
// Kernel B: rank channels per batch. Descending rank (ties -> lower index
// first) selects top-K_TOP in jax top_k order; ascending rank selects the
// low-K_LOW set in top_k(-pooled) order. Ranks are unique -> no collisions.
// ---------------------------------------------------------------------------
__global__ void se_rank_kernel(const float* __restrict__ pooled,
                               int* __restrict__ perm_top,
                               int* __restrict__ perm_low) {
    __shared__ float pv[C_];
    const int b = blockIdx.x;
    const int c = threadIdx.x;                       // 512 threads
    pv[c] = pooled[b * C_ + c];
    __syncthreads();
    const float v = pv[c];
    int rd = 0, ra = 0;
    for (int c2 = 0; c2 < C_; ++c2) {
        const float u = pv[c2];
        const bool tie_lo = (u == v) && (c2 < c);
        rd += ((u > v) || tie_lo) ? 1 : 0;
        ra += ((u < v) || tie_lo) ? 1 : 0;
    }
    if (rd < KTOP_) perm_top[b * KTOP_ + rd] = c;
    if (ra < KLOW_) perm_low[b * KLOW_ + ra] = c;
}

// ---------------------------------------------------------------------------
// Kernel C: bandwidth copy of top channels into output slots (float4).
// ---------------------------------------------------------------------------
__global__ void se_gather_kernel(const float* __restrict__ x,
                                 const int* __restrict__ perm_top,
                                 float* __restrict__ out) {
    const int o = blockIdx.x;                        // 0..KTOP_-1
    const int b = blockIdx.y;
    const int src = perm_top[b * KTOP_ + o];
    const float4* s = (const float4*)(x   + (size_t)(b * C_ + src) * HW_);
    float4*       d = (float4*)      (out + (size_t)(b * C_ + o)   * HW_);
    for (int i = threadIdx.x; i < HW_ / 4; i += 256) d[i] = s[i];
}

// ---------------------------------------------------------------------------
// Kernel D: fused gather + 1x1 conv via V_WMMA_F32_16X16X4_F32.
// Block = (pixel chunk of 112, batch). 224 threads = 7 waves (wave32).
// LDS: Wl[112][104] (padded weights), Xl[104][112] (gathered low channels).
// Wave w computes N-tile w for all 7 M-tiles; 26 K-steps of 4.
// ---------------------------------------------------------------------------
__global__ void se_fusion_wmma_kernel(const float* __restrict__ x,
                                      const float* __restrict__ w,
                                      const float* __restrict__ bias,
                                      const int* __restrict__ perm_low,
                                      float* __restrict__ out) {
    extern __shared__ float smem[];
    float* Wl = smem;                 // MP_ * KP_ floats (row-major, M x K)
    float* Xl = smem + MP_ * KP_;     // KP_ * NCH_ floats (row-major, K x N)

    const int b   = blockIdx.y;
    const int p0  = blockIdx.x * NCH_;
    const int tid = threadIdx.x;      // 0..223
    const int* pl = perm_low + b * KLOW_;

    // Stage zero-padded weights: Wl[m][k] = w_fusion[m][k]
    for (int i = tid; i < MP_ * KP_; i += 224) {
        const int m = i / KP_;
        const int k = i - m * KP_;
        Wl[i] = (m < KLOW_ && k < KLOW_) ? w[m * KLOW_ + k] : 0.f;
    }
    // Stage gathered low-channel pixel tile: Xl[c][p] = x[b, perm_low[c], p0+p]
    for (int i = tid; i < KP_ * NCH_; i += 224) {
        const int c = i / NCH_;
        const int p = i - c * NCH_;
        float v = 0.f;
        if (c < KLOW_) {
            const int src = pl[c];
            v = x[(size_t)(b * C_ + src) * HW_ + p0 + p];
        }
        Xl[i] = v;
    }
    __syncthreads();

    const int wave = tid >> 5;        // 0..6  -> N tile
    const int lane = tid & 31;
    const int hi   = lane >> 4;       // half-wave selects K pair / M high rows
    const int ln   = lane & 15;
    const int n    = wave * 16 + ln;  // pixel column within chunk

    v8f acc[7];
    #pragma unroll
    for (int mt = 0; mt < 7; ++mt) acc[mt] = (v8f){0.f,0.f,0.f,0.f,0.f,0.f,0.f,0.f};

    for (int ks = 0; ks < KSTEPS_; ++ks) {
        const int k0 = ks * 4;
        // B fragment (4x16): lanes 0-15 hold K={k0,k0+1}, lanes 16-31 K={k0+2,k0+3}
        v2f bf;
        bf.x = Xl[(k0 + 2 * hi)     * NCH_ + n];
        bf.y = Xl[(k0 + 2 * hi + 1) * NCH_ + n];
        #pragma unroll
        for (int mt = 0; mt < 7; ++mt) {
            // A fragment (16x4): lane row = ln, cols {k0+2*hi, k0+2*hi+1}
            const v2f af = *(const v2f*)(&Wl[(mt * 16 + ln) * KP_ + k0 + 2 * hi]);
            acc[mt] = __builtin_amdgcn_wmma_f32_16x16x4_f32(
                /*neg_a=*/false, af, /*neg_b=*/false, bf,
                /*c_mod=*/(short)0, acc[mt], /*reuse_a=*/false, /*reuse_b=*/false);
        }
    }

    // D layout: VGPR r, lane -> row m = mt*16 + 8*hi + r, col n. Add bias, store.
    #pragma unroll
    for (int mt = 0; mt < 7; ++mt) {
        #pragma unroll
        for (int r = 0; r < 8; ++r) {
            const int m = mt * 16 + hi * 8 + r;
            if (m < KLOW_) {
                out[(size_t)(b * C_ + KTOP_ + m) * HW_ + p0 + n] = acc[mt][r] + bias[m];
            }
        }
    }
}

// ---------------------------------------------------------------------------
extern "C" void kernel_launch(void* const* d_in, const int* in_sizes, int n_in,
                              void* d_out, int out_size, void* d_ws, size_t ws_size,
                              hipStream_t stream) {
    const float* x    = (const float*)d_in[0];   // [32,512,56,56]
    const float* w    = (const float*)d_in[1];   // [103,103]
    const float* bias = (const float*)d_in[2];   // [103]
    float* out = (float*)d_out;                  // [32,512,56,56]

    // Workspace carve-out
    float* pooled   = (float*)d_ws;                    // B*C floats
    int*   perm_top = (int*)(pooled + B_ * C_);        // B*KTOP ints
    int*   perm_low = (int*)(perm_top + B_ * KTOP_);   // B*KLOW ints

    se_pool_kernel<<<B_ * C_, 256, 0, stream>>>(x, pooled);
    se_rank_kernel<<<B_, C_, 0, stream>>>(pooled, perm_top, perm_low);

    dim3 gGather(KTOP_, B_);
    se_gather_kernel<<<gGather, 256, 0, stream>>>(x, perm_top, out);

    dim3 gFusion(HW_ / NCH_, B_);                      // 28 x 32
    const size_t ldsBytes = (size_t)(MP_ * KP_ + KP_ * NCH_) * sizeof(float); // ~93 KB
    se_fusion_wmma_kernel<<<gFusion, 224, ldsBytes, stream>>>(x, w, bias, perm_low, out);
}